// FastSpeech2_21268678049845
// MI455X (gfx1250) — compile-verified
//
#include <hip/hip_runtime.h>
#include <hip/hip_bf16.h>

typedef float v2f __attribute__((ext_vector_type(2)));
typedef float v8f __attribute__((ext_vector_type(8)));

#define BATCH   4
#define L_SRC   96
#define D_MODEL 384
#define N_HEADS 2
#define HEAD_D  192
#define D_FF    1536
#define N_MELS  80
#define T_MEL   2845

__device__ __forceinline__ int imin(int a, int b) { return a < b ? a : b; }

// D = A(16x4 f32) * B(4x16 f32) + C(16x16 f32)  — CDNA5 WMMA
__device__ __forceinline__ v8f wmma4(v2f a, v2f b, v8f c) {
  return __builtin_amdgcn_wmma_f32_16x16x4_f32(false, a, false, b, (short)0, c,
                                               false, false);
}

// sinusoidal positional encoding value, matches reference _pos_enc for d=384
__device__ __forceinline__ float pe_val(int pos, int d) {
  int k2 = (d >> 1) * 2;                       // even index 0,2,...,382
  float div = expf((float)k2 * (-9.210340371976184f / (float)D_MODEL));
  float x = (float)pos * div;
  return (d & 1) ? cosf(x) : sinf(x);
}

// ---------------------------------------------------------------- embedding
__global__ void embed_kernel(const int* __restrict__ src,
                             const float* __restrict__ emb,
                             float* __restrict__ x) {
  int row = blockIdx.x;            // b*L_SRC + s
  int s = row % L_SRC;
  int d = threadIdx.x;
  int tok = src[row];
  x[(size_t)row * D_MODEL + d] = emb[(size_t)tok * D_MODEL + d] + pe_val(s, d);
}

// ------------------------------------------------------- length regulator
__global__ void lenreg_kernel(const float* __restrict__ xe,
                              float* __restrict__ y) {
  int row = blockIdx.x;            // b*T_MEL + t
  int b = row / T_MEL, t = row % T_MEL;
  int idx; float alpha;
  if (t < 10)        { idx = 0;  alpha = (float)t / 10.0f; }
  else if (t < 2830) { int u = t - 10; idx = 1 + u / 30; alpha = (float)(u % 30) / 30.0f; }
  else               { idx = L_SRC - 1; alpha = 0.0f; }
  int nxt = imin(idx + 1, L_SRC - 1);
  int d = threadIdx.x;
  float va = xe[((size_t)b * L_SRC + idx) * D_MODEL + d];
  float vb = xe[((size_t)b * L_SRC + nxt) * D_MODEL + d];
  y[(size_t)row * D_MODEL + d] = (1.0f - alpha) * va + alpha * vb + pe_val(t, d);
}

// -------------------------------------------------- residual + layer norm
// Y[row] = LN(X[row] + (T? T[row]:0)) * g + b     (block = 384 threads)
__global__ void add_ln_kernel(const float* __restrict__ X,
                              const float* __restrict__ T,
                              const float* __restrict__ g,
                              const float* __restrict__ b,
                              float* __restrict__ Y, int M) {
  int row = blockIdx.x;
  int d = threadIdx.x;
  size_t o = (size_t)row * D_MODEL + d;
  float v = X[o] + (T ? T[o] : 0.0f);
  float s1 = v, s2 = v * v;
#pragma unroll
  for (int off = 1; off < 32; off <<= 1) {
    s1 += __shfl_xor(s1, off);
    s2 += __shfl_xor(s2, off);
  }
  __shared__ float red[24];
  __shared__ float mv[2];
  int w = d >> 5;
  if ((d & 31) == 0) { red[w] = s1; red[12 + w] = s2; }
  __syncthreads();
  if (d == 0) {
    float a = 0.f, c = 0.f;
#pragma unroll
    for (int i = 0; i < 12; ++i) { a += red[i]; c += red[12 + i]; }
    float mean = a / (float)D_MODEL;
    float var = c / (float)D_MODEL - mean * mean;
    mv[0] = mean;
    mv[1] = rsqrtf(var + 1e-5f);
  }
  __syncthreads();
  Y[o] = (v - mv[0]) * mv[1] * g[d] + b[d];
}

// ------------------------------------------------------------- WMMA GEMM
// Y[M x N] = X[M x K] @ W[K x N] + bias.  One wave per 16 x (16*NT) strip:
// the A fragment is reused across NT column tiles and NT independent WMMA
// chains per k-step hide load latency.
// MODE 0: plain store.  MODE 2: mel transpose store out[b][n][t], row=b*S+t.
template <int MODE, int NT>
__global__ __launch_bounds__(32, 1)
void gemm_kernel(const float* __restrict__ X,
                 const float* __restrict__ W,
                 const float* __restrict__ bias,
                 float* __restrict__ Y,
                 int M, int K, int N, int S) {
  int lane = threadIdx.x;
  int half = lane >> 4, l15 = lane & 15;
  int m0 = blockIdx.x * 16, n0 = blockIdx.y * 16 * NT;
  int arow = imin(m0 + l15, M - 1);
  const float* xr = X + (size_t)arow * K;
  v8f oz = {};
  v8f acc[NT];
#pragma unroll
  for (int t = 0; t < NT; ++t) acc[t] = oz;
  for (int k0 = 0; k0 < K; k0 += 4) {
    int ka = k0 + 2 * half;
    v2f a; a.x = xr[ka]; a.y = xr[ka + 1];
    const float* w0 = W + (size_t)ka * N + n0 + l15;
#pragma unroll
    for (int t = 0; t < NT; ++t) {
      v2f b; b.x = w0[t * 16]; b.y = w0[N + t * 16];
      acc[t] = wmma4(a, b, acc[t]);
    }
  }
#pragma unroll
  for (int t = 0; t < NT; ++t) {
    int bcol = n0 + t * 16 + l15;
    float bn = bias[bcol];
#pragma unroll
    for (int r = 0; r < 8; ++r) {
      int row = m0 + r + 8 * half;
      if (row < M) {
        float val = acc[t][r] + bn;
        if (MODE == 2) {
          int bb = row / S, tt = row - bb * S;
          Y[((size_t)bb * N + bcol) * S + tt] = val;
        } else {
          Y[(size_t)row * N + bcol] = val;
        }
      }
    }
  }
}

// ------------------------------------------------- flash attention (WMMA)
// grid = (ceil(S/16), N_HEADS, BATCH), block = 32 (one wave).
// Processes 32 keys per outer iteration: two 16x16 score tiles share the
// softmax butterfly reductions, halving LDS round-trips per key, and each
// Q A-fragment read from LDS feeds two QK^T WMMAs.
__global__ __launch_bounds__(32, 1)
void attn_kernel(const float* __restrict__ Q,
                 const float* __restrict__ Kb,
                 const float* __restrict__ Vb,
                 float* __restrict__ O, int S) {
  __shared__ float qs[16 * HEAD_D];
  __shared__ float ps[16 * 32];
  int lane = threadIdx.x;
  int half = lane >> 4, l15 = lane & 15;
  int qt = blockIdx.x, hh = blockIdx.y, bb = blockIdx.z;
  const float sc_qk = rsqrtf((float)HEAD_D);

  for (int i = lane; i < 16 * HEAD_D; i += 32) {
    int r = i / HEAD_D, c = i - r * HEAD_D;
    int srow = imin(qt * 16 + r, S - 1);
    qs[i] = Q[((size_t)bb * S + srow) * D_MODEL + hh * HEAD_D + c] * sc_qk;
  }
  __syncthreads();

  v8f oz = {};
  v8f oac[12];
#pragma unroll
  for (int t = 0; t < 12; ++t) oac[t] = oz;
  float mrun[8], lrun[8];
#pragma unroll
  for (int r = 0; r < 8; ++r) { mrun[r] = -1e30f; lrun[r] = 0.0f; }

  for (int kt = 0; kt < S; kt += 32) {
    // ---- scores: two 16x16 tiles = Qtile @ K^T for keys [kt, kt+32)
    v8f s0 = oz, s1 = oz;
    int kr0 = imin(kt + l15, S - 1);
    int kr1 = imin(kt + 16 + l15, S - 1);
    const float* krow0 = Kb + ((size_t)bb * S + kr0) * D_MODEL + hh * HEAD_D;
    const float* krow1 = Kb + ((size_t)bb * S + kr1) * D_MODEL + hh * HEAD_D;
#pragma unroll 8
    for (int k0 = 0; k0 < HEAD_D; k0 += 4) {
      int ka = k0 + 2 * half;
      v2f a; a.x = qs[l15 * HEAD_D + ka]; a.y = qs[l15 * HEAD_D + ka + 1];
      v2f b0; b0.x = krow0[ka]; b0.y = krow0[ka + 1];
      v2f b1; b1.x = krow1[ka]; b1.y = krow1[ka + 1];
      s0 = wmma4(a, b0, s0);
      s1 = wmma4(a, b1, s1);
    }
    float kv0 = ((kt + l15) < S) ? 0.0f : -1e30f;
    float kv1 = ((kt + 16 + l15) < S) ? 0.0f : -1e30f;
    // ---- online softmax over the 32 keys; butterflies batched for ILP
    float sv0[8], sv1[8], mx[8];
#pragma unroll
    for (int r = 0; r < 8; ++r) {
      sv0[r] = s0[r] + kv0;
      sv1[r] = s1[r] + kv1;
      mx[r] = fmaxf(sv0[r], sv1[r]);
    }
#pragma unroll
    for (int off = 1; off < 16; off <<= 1) {
      float tmp[8];
#pragma unroll
      for (int r = 0; r < 8; ++r) tmp[r] = __shfl_xor(mx[r], off);
#pragma unroll
      for (int r = 0; r < 8; ++r) mx[r] = fmaxf(mx[r], tmp[r]);
    }
    float p0[8], p1[8], su[8], scl[8];
#pragma unroll
    for (int r = 0; r < 8; ++r) {
      float mnew = fmaxf(mrun[r], mx[r]);
      scl[r] = expf(mrun[r] - mnew);
      mrun[r] = mnew;
      p0[r] = expf(sv0[r] - mnew);
      p1[r] = expf(sv1[r] - mnew);
      su[r] = p0[r] + p1[r];
    }
#pragma unroll
    for (int off = 1; off < 16; off <<= 1) {
      float tmp[8];
#pragma unroll
      for (int r = 0; r < 8; ++r) tmp[r] = __shfl_xor(su[r], off);
#pragma unroll
      for (int r = 0; r < 8; ++r) su[r] += tmp[r];
    }
#pragma unroll
    for (int r = 0; r < 8; ++r) lrun[r] = lrun[r] * scl[r] + su[r];
#pragma unroll
    for (int t = 0; t < 12; ++t)
#pragma unroll
      for (int r = 0; r < 8; ++r) oac[t][r] *= scl[r];
    // ---- P (C/D layout) -> LDS (16x32 row-major) -> A-fragment layout
    __syncthreads();
#pragma unroll
    for (int r = 0; r < 8; ++r) {
      ps[(r + 8 * half) * 32 + l15] = p0[r];
      ps[(r + 8 * half) * 32 + 16 + l15] = p1[r];
    }
    __syncthreads();
    // ---- O += P @ V over 32 keys
#pragma unroll
    for (int kk = 0; kk < 32; kk += 4) {
      int ka = kk + 2 * half;
      v2f a; a.x = ps[l15 * 32 + ka]; a.y = ps[l15 * 32 + ka + 1];
      int vr0 = imin(kt + ka, S - 1);
      int vr1 = imin(kt + ka + 1, S - 1);
      const float* v0 = Vb + ((size_t)bb * S + vr0) * D_MODEL + hh * HEAD_D;
      const float* v1 = Vb + ((size_t)bb * S + vr1) * D_MODEL + hh * HEAD_D;
#pragma unroll
      for (int t = 0; t < 12; ++t) {
        v2f b; b.x = v0[t * 16 + l15]; b.y = v1[t * 16 + l15];
        oac[t] = wmma4(a, b, oac[t]);
      }
    }
  }
#pragma unroll
  for (int t = 0; t < 12; ++t)
#pragma unroll
    for (int r = 0; r < 8; ++r) {
      int row = qt * 16 + r + 8 * half;
      if (row < S)
        O[((size_t)bb * S + row) * D_MODEL + hh * HEAD_D + t * 16 + l15] =
            oac[t][r] / lrun[r];
    }
}

// ------------------------------------------------------ fused FFN (WMMA)
// Y = relu(X @ w1 + c1) @ w2 + c2, hidden staged through LDS in 256-chunks.
// grid = ceil(M/16), block = 64 (two waves).  Wave w computes half of the
// hidden chunk (nt = w, w+2, ...) and owns 12 of the 24 output tiles, so
// accumulator pressure is 96 VGPRs/wave and the LDS h-chunk is shared.
__global__ __launch_bounds__(64, 1)
void ffn_kernel(const float* __restrict__ X,
                const float* __restrict__ w1,
                const float* __restrict__ c1,
                const float* __restrict__ w2,
                const float* __restrict__ c2,
                float* __restrict__ Y, int M) {
  const int CH = 256;
  __shared__ float hs[16 * CH];
  int lane = threadIdx.x & 31;
  int wave = threadIdx.x >> 5;
  int half = lane >> 4, l15 = lane & 15;
  int m0 = blockIdx.x * 16;
  int arow = imin(m0 + l15, M - 1);
  const float* xr = X + (size_t)arow * D_MODEL;

  v8f oz = {};
  v8f acc[12];
#pragma unroll
  for (int t = 0; t < 12; ++t) acc[t] = oz;

  for (int c0 = 0; c0 < D_FF; c0 += CH) {
    __syncthreads();
    // h chunk = relu(X @ w1[:, c0:c0+CH] + c1); waves interleave column tiles
    for (int nt = wave; nt < CH / 16; nt += 2) {
      v8f h = oz;
      int col = c0 + nt * 16 + l15;
#pragma unroll 8
      for (int k0 = 0; k0 < D_MODEL; k0 += 4) {
        int ka = k0 + 2 * half;
        v2f a; a.x = xr[ka]; a.y = xr[ka + 1];
        v2f b; b.x = w1[(size_t)ka * D_FF + col];
        b.y = w1[(size_t)(ka + 1) * D_FF + col];
        h = wmma4(a, b, h);
      }
      float bb = c1[col];
#pragma unroll
      for (int r = 0; r < 8; ++r)
        hs[(r + 8 * half) * CH + nt * 16 + l15] = fmaxf(h[r] + bb, 0.0f);
    }
    __syncthreads();
    // acc += h_chunk @ w2[c0:c0+CH, wave's 192-column half]
    for (int kk = 0; kk < CH; kk += 4) {
      int ka = kk + 2 * half;
      v2f a; a.x = hs[l15 * CH + ka]; a.y = hs[l15 * CH + ka + 1];
      int g0 = c0 + ka;
      const float* w2a = w2 + (size_t)g0 * D_MODEL + wave * 192;
      const float* w2b = w2a + D_MODEL;
#pragma unroll
      for (int t = 0; t < 12; ++t) {
        v2f b; b.x = w2a[t * 16 + l15]; b.y = w2b[t * 16 + l15];
        acc[t] = wmma4(a, b, acc[t]);
      }
    }
  }
#pragma unroll
  for (int t = 0; t < 12; ++t) {
    int col = wave * 192 + t * 16 + l15;
    float bb = c2[col];
#pragma unroll
    for (int r = 0; r < 8; ++r) {
      int row = m0 + r + 8 * half;
      if (row < M) Y[(size_t)row * D_MODEL + col] = acc[t][r] + bb;
    }
  }
}

// =========================================================== host side
struct StackP {
  const float *wq, *bq, *wk, *bk, *wv, *bv, *wo, *bo;
  const float *w1, *c1, *w2, *c2, *g1, *b1, *g2, *b2;
};

static void fill_stack(StackP& p, void* const* d_in, int base, bool sorted) {
  auto F = [&](int i) { return (const float*)d_in[base + i]; };
  if (!sorted) {  // insertion order: wq,bq,wk,bk,wv,bv,wo,bo,w1,c1,w2,c2,g1,b1,g2,b2
    p.wq = F(0);  p.bq = F(1);  p.wk = F(2);  p.bk = F(3);
    p.wv = F(4);  p.bv = F(5);  p.wo = F(6);  p.bo = F(7);
    p.w1 = F(8);  p.c1 = F(9);  p.w2 = F(10); p.c2 = F(11);
    p.g1 = F(12); p.b1 = F(13); p.g2 = F(14); p.b2 = F(15);
  } else {        // sorted keys: b1,b2,bk,bo,bq,bv,c1,c2,g1,g2,w1,w2,wk,wo,wq,wv
    p.b1 = F(0);  p.b2 = F(1);  p.bk = F(2);  p.bo = F(3);
    p.bq = F(4);  p.bv = F(5);  p.c1 = F(6);  p.c2 = F(7);
    p.g1 = F(8);  p.g2 = F(9);  p.w1 = F(10); p.w2 = F(11);
    p.wk = F(12); p.wo = F(13); p.wq = F(14); p.wv = F(15);
  }
}

// strides between stacked layers (elements)
#define SW  (D_MODEL * D_MODEL)   // wq/wk/wv/wo
#define SB  (D_MODEL)             // biases / ln vectors / c2
#define SW1 (D_MODEL * D_FF)      // w1
#define SC1 (D_FF)                // c1
#define SW2 (D_FF * D_MODEL)      // w2

static void run_block(const StackP& P, int l, float* x, float* q, float* k,
                      float* v, float* o, float* t, int S, int M,
                      hipStream_t stream) {
  dim3 w32(32);
  int mt = (M + 15) / 16;
  dim3 gQ(mt, D_MODEL / 64);     // NT=4 column strips
  gemm_kernel<0, 4><<<gQ, w32, 0, stream>>>(x, P.wq + (size_t)l * SW, P.bq + l * SB, q, M, D_MODEL, D_MODEL, S);
  gemm_kernel<0, 4><<<gQ, w32, 0, stream>>>(x, P.wk + (size_t)l * SW, P.bk + l * SB, k, M, D_MODEL, D_MODEL, S);
  gemm_kernel<0, 4><<<gQ, w32, 0, stream>>>(x, P.wv + (size_t)l * SW, P.bv + l * SB, v, M, D_MODEL, D_MODEL, S);
  dim3 gA((S + 15) / 16, N_HEADS, BATCH);
  attn_kernel<<<gA, w32, 0, stream>>>(q, k, v, o, S);
  gemm_kernel<0, 4><<<gQ, w32, 0, stream>>>(o, P.wo + (size_t)l * SW, P.bo + l * SB, t, M, D_MODEL, D_MODEL, S);
  add_ln_kernel<<<dim3(M), dim3(D_MODEL), 0, stream>>>(x, t, P.g1 + l * SB, P.b1 + l * SB, x, M);
  ffn_kernel<<<dim3(mt), dim3(64), 0, stream>>>(x, P.w1 + (size_t)l * SW1, P.c1 + (size_t)l * SC1,
                                                P.w2 + (size_t)l * SW2, P.c2 + l * SB, t, M);
  add_ln_kernel<<<dim3(M), dim3(D_MODEL), 0, stream>>>(x, t, P.g2 + l * SB, P.b2 + l * SB, x, M);
}

extern "C" void kernel_launch(void* const* d_in, const int* in_sizes, int n_in,
                              void* d_out, int out_size, void* d_ws, size_t ws_size,
                              hipStream_t stream) {
  // Detect harness flattening order: insertion (src first, 384 ints) vs
  // pytree-sorted (dec_params.b1 first, 1536 floats).
  bool sorted = (in_sizes[0] != BATCH * L_SRC);

  const int* src; const float *emb, *ln_g, *ln_b, *mel_w, *mel_b;
  StackP enc, dec;
  if (!sorted) {
    src = (const int*)d_in[0];
    emb = (const float*)d_in[1];
    fill_stack(enc, d_in, 2, false);
    fill_stack(dec, d_in, 18, false);
    ln_g = (const float*)d_in[34]; ln_b = (const float*)d_in[35];
    mel_w = (const float*)d_in[36]; mel_b = (const float*)d_in[37];
  } else {  // top-level sorted: dec_params, embedding, enc_params, ln_b, ln_g, mel_b, mel_w, src
    fill_stack(dec, d_in, 0, true);
    emb = (const float*)d_in[16];
    fill_stack(enc, d_in, 17, true);
    ln_b = (const float*)d_in[33]; ln_g = (const float*)d_in[34];
    mel_b = (const float*)d_in[35]; mel_w = (const float*)d_in[36];
    src = (const int*)d_in[37];
  }

  const int ME = BATCH * L_SRC;        // 384 encoder rows
  const int MD = BATCH * T_MEL;        // 11380 decoder rows
  const size_t R = (size_t)MD * D_MODEL;  // per-buffer floats
  float* ws = (float*)d_ws;
  float* f0 = ws;          // encoder x, later decoder q/t scratch
  float* f1 = ws + R;      // decoder x
  float* f2 = ws + 2 * R;  // k
  float* f3 = ws + 3 * R;  // v
  float* f4 = ws + 4 * R;  // attn out

  dim3 w32(32);

  // ---- encoder ----
  embed_kernel<<<dim3(ME), dim3(D_MODEL), 0, stream>>>(src, emb, f0);
  for (int l = 0; l < 4; ++l)
    run_block(enc, l, f0, f1, f2, f3, f4, f1, L_SRC, ME, stream);

  // ---- length regulator + decoder PE ----
  lenreg_kernel<<<dim3(MD), dim3(D_MODEL), 0, stream>>>(f0, f1);

  // ---- decoder ----
  for (int l = 0; l < 4; ++l)
    run_block(dec, l, f1, f0, f2, f3, f4, f0, T_MEL, MD, stream);

  // ---- final LN + mel projection (transposed store, N=80 in one strip) ----
  add_ln_kernel<<<dim3(MD), dim3(D_MODEL), 0, stream>>>(f1, nullptr, ln_g, ln_b, f1, MD);
  dim3 gM((MD + 15) / 16, 1);
  gemm_kernel<2, 5><<<gM, w32, 0, stream>>>(f1, mel_w, mel_b, (float*)d_out,
                                            MD, D_MODEL, N_MELS, T_MEL);
}